// ACPLoss_39487929319787
// MI455X (gfx1250) — compile-verified
//
#include <hip/hip_runtime.h>
#include <math.h>

typedef float v2f __attribute__((ext_vector_type(2)));
typedef float v8f __attribute__((ext_vector_type(8)));

#define B_ROWS 512
#define D_DIM  512
#define C_COLS 100000
#define S_SCALE 64.0f
#define INV_S   (1.0f/64.0f)
#define COS_M_F   0.796083798549056f
#define SIN_M_F   0.605186405736040f
#define THRESH_F (-0.796083798549056f)
#define MM_F      0.393371163728426f
#define WEIGHT_F  0.003f

#define BN 32                 // columns per block
#define KC 8                  // k-chunk (2 wmma k-steps)
#define NCHUNK (D_DIM / KC)   // 64

// ---------------- K1: inverse L2 norm of each embedding row ----------------
__global__ __launch_bounds__(256) void k_row_norms(const float* __restrict__ emb,
                                                   float* __restrict__ rinv) {
  __shared__ float red[256];
  const int b = blockIdx.x;
  const int t = threadIdx.x;
  const float* row = emb + (size_t)b * D_DIM;
  const float x0 = row[t];
  const float x1 = row[t + 256];
  red[t] = x0 * x0 + x1 * x1;
  __syncthreads();
  for (int s = 128; s > 0; s >>= 1) {
    if (t < s) red[t] += red[t + s];
    __syncthreads();
  }
  if (t == 0) rinv[b] = 1.0f / sqrtf(red[0]);
}

// ---------------- K2: fused column-norm + GEMM (WMMA f32 16x16x4) ----------
// Block: 256 threads (8 waves). Covers all 512 rows x 32 columns, so the
// weight matrix streams from HBM exactly once; column sum-of-squares is
// accumulated during the same pass. Double-buffered LDS staging: global
// loads for chunk i+1 are issued before computing chunk i.
__global__ __launch_bounds__(256) void k_gemm(const float* __restrict__ W,
                                              const float* __restrict__ emb,
                                              const float* __restrict__ rinv,
                                              float* __restrict__ out2) {
  __shared__ float As[2][D_DIM * 12];   // [buf][512 rows][KC=8 padded to 12]
  __shared__ float Bs[2][KC * 33];      // [buf][8 k][32 cols padded to 33]
  __shared__ float rI[512];
  __shared__ float red[8 * BN];
  __shared__ float cInv[BN];

  const int t     = threadIdx.x;
  const int n0    = blockIdx.x * BN;
  const int wave  = t >> 5;
  const int lane  = t & 31;
  const int half  = lane >> 4;       // selects K-pair (A/B) or M+8 (C/D)
  const int lp    = lane & 15;
  const int mbase = (wave >> 1) * 128;
  const int nbase = (wave & 1) * 16;

  // stage inverse row norms (needed by the store phase of every chunk)
  rI[t]       = rinv[t];
  rI[t + 256] = rinv[t + 256];
  __syncthreads();

  v8f acc[8];
  #pragma unroll
  for (int i = 0; i < 8; ++i)
    #pragma unroll
    for (int j = 0; j < 8; ++j) acc[i][j] = 0.0f;

  float csum = 0.0f;                 // per-thread partial column sum-of-squares
  const int kr    = t >> 5;          // 0..7   : k-row of B chunk
  const int cl    = t & 31;          // 0..31  : column of B chunk
  const int arow  = t >> 1;          // 0..127 : A-chunk row within pass
  const int ahalf = t & 1;           // which float4 of the 8-wide k slice

  // ---- prologue: load + stage chunk 0 into buffer 0
  float  bv;
  float4 av[4];
  bv = W[(size_t)kr * C_COLS + (n0 + cl)];
  #pragma unroll
  for (int p = 0; p < 4; ++p) {
    const int r = p * 128 + arow;
    av[p] = *reinterpret_cast<const float4*>(emb + (size_t)r * D_DIM + ahalf * 4);
  }
  csum += bv * bv;
  Bs[0][kr * 33 + cl] = bv;
  #pragma unroll
  for (int p = 0; p < 4; ++p) {
    const int r = p * 128 + arow;
    const float s = rI[r];
    float4 g; g.x = av[p].x * s; g.y = av[p].y * s; g.z = av[p].z * s; g.w = av[p].w * s;
    *reinterpret_cast<float4*>(&As[0][r * 12 + ahalf * 4]) = g;
  }
  __syncthreads();

  int cur = 0;
  for (int i = 0; i < NCHUNK; ++i) {
    const bool has_next = (i + 1) < NCHUNK;
    const int  kc_next  = (i + 1) * KC;
    // ---- issue global loads for chunk i+1 (overlap with WMMA burst below)
    if (has_next) {
      bv = W[(size_t)(kc_next + kr) * C_COLS + (n0 + cl)];
      #pragma unroll
      for (int p = 0; p < 4; ++p) {
        const int r = p * 128 + arow;
        av[p] = *reinterpret_cast<const float4*>(emb + (size_t)r * D_DIM + kc_next + ahalf * 4);
      }
    }
    // ---- compute chunk i from LDS buffer `cur`
    const float* Ab = &As[cur][0];
    const float* Bb = &Bs[cur][0];
    #pragma unroll
    for (int ks = 0; ks < 2; ++ks) {
      const int krow = ks * 4 + half * 2;          // B: row K = v + 2*half
      v2f bf;
      bf.x = Bb[krow * 33 + nbase + lp];
      bf.y = Bb[(krow + 1) * 33 + nbase + lp];
      #pragma unroll
      for (int mt = 0; mt < 8; ++mt) {
        const int m = mbase + mt * 16 + lp;        // A: row M = lane%16
        const v2f af = *reinterpret_cast<const v2f*>(&Ab[m * 12 + ks * 4 + half * 2]);
        acc[mt] = __builtin_amdgcn_wmma_f32_16x16x4_f32(
            /*neg_a=*/false, af, /*neg_b=*/false, bf,
            /*c_mod=*/(short)0, acc[mt], /*reuse_a=*/false, /*reuse_b=*/false);
      }
    }
    // ---- stage chunk i+1 into the alternate buffer
    if (has_next) {
      csum += bv * bv;
      Bs[cur ^ 1][kr * 33 + cl] = bv;
      #pragma unroll
      for (int p = 0; p < 4; ++p) {
        const int r = p * 128 + arow;
        const float s = rI[r];
        float4 g; g.x = av[p].x * s; g.y = av[p].y * s; g.z = av[p].z * s; g.w = av[p].w * s;
        *reinterpret_cast<float4*>(&As[cur ^ 1][r * 12 + ahalf * 4]) = g;
      }
    }
    __syncthreads();
    cur ^= 1;
  }

  // --- reduce column sum-of-squares -> inverse column norms
  red[kr * BN + cl] = csum;
  __syncthreads();
  if (t < BN) {
    float s = 0.0f;
    #pragma unroll
    for (int i = 0; i < 8; ++i) s += red[i * BN + t];
    cInv[t] = 1.0f / sqrtf(s);
  }
  __syncthreads();

  // --- epilogue: clamp(cos) * S  -> origin_cos*S region
  const float rc  = cInv[nbase + lp];
  const int   col = n0 + nbase + lp;
  #pragma unroll
  for (int mt = 0; mt < 8; ++mt) {
    #pragma unroll
    for (int vv = 0; vv < 8; ++vv) {
      const int row = mbase + mt * 16 + half * 8 + vv;   // C/D: M = v + 8*half
      const float cosv = fminf(fmaxf(acc[mt][vv] * rc, -1.0f), 1.0f);
      out2[(size_t)row * C_COLS + col] = cosv * S_SCALE;
    }
  }
}

// ---------------- K3: target logits, t update, center loss -----------------
__global__ __launch_bounds__(512) void k_target(const float* __restrict__ out2,
                                                const int* __restrict__ label,
                                                const float* __restrict__ t_in,
                                                float* __restrict__ ctm_a,
                                                float* __restrict__ ftl_a,
                                                float* __restrict__ tnew_p,
                                                float* __restrict__ scalar_out) {
  __shared__ float r1[512];
  __shared__ float r2[512];
  const int b = threadIdx.x;
  const int lb = label[b];
  const float tl = out2[(size_t)b * C_COLS + lb] * INV_S;   // exact (S = 2^6)
  const float sn  = sqrtf(fmaxf(0.0f, 1.0f - tl * tl));
  const float ctm = tl * COS_M_F - sn * SIN_M_F;
  const float ftl = (tl > THRESH_F) ? ctm : (tl - MM_F);
  ctm_a[b] = ctm;
  ftl_a[b] = ftl;
  const float ac = acosf(tl);
  r1[b] = tl;
  r2[b] = ac * sqrtf(ac);                                   // acos^1.5
  __syncthreads();
  for (int s = 256; s > 0; s >>= 1) {
    if (b < s) { r1[b] += r1[b + s]; r2[b] += r2[b + s]; }
    __syncthreads();
  }
  if (b == 0) {
    tnew_p[0]     = (r1[0] * (1.0f / 512.0f)) * 0.01f + 0.99f * t_in[0];
    scalar_out[0] = S_SCALE * WEIGHT_F * (r2[0] * (1.0f / 512.0f));
  }
}

// ---------------- K4: hard-example reweight + label scatter ----------------
__global__ __launch_bounds__(256) void k_epilogue(const float* __restrict__ out2,
                                                  const int* __restrict__ label,
                                                  const float* __restrict__ ctm_a,
                                                  const float* __restrict__ ftl_a,
                                                  const float* __restrict__ tnew_p,
                                                  float* __restrict__ out) {
  const size_t gid = (size_t)blockIdx.x * blockDim.x + threadIdx.x;
  const size_t i4  = gid * 4;
  const int b  = (int)(i4 / C_COLS);               // C divisible by 4: row-aligned
  const int c0 = (int)(i4 - (size_t)b * C_COLS);
  const float tn  = tnew_p[0];
  const float ctm = ctm_a[b];
  const float ftl = ftl_a[b];
  const int   lb  = label[b];
  const float4 o = *reinterpret_cast<const float4*>(out2 + i4);
  float r[4] = {o.x, o.y, o.z, o.w};
  #pragma unroll
  for (int j = 0; j < 4; ++j) {
    const float cosv = r[j] * INV_S;
    float val = (cosv > ctm) ? cosv * (tn + cosv) : cosv;
    if (c0 + j == lb) val = ftl;
    r[j] = val * S_SCALE;
  }
  float4 res; res.x = r[0]; res.y = r[1]; res.z = r[2]; res.w = r[3];
  *reinterpret_cast<float4*>(out + i4) = res;
}

extern "C" void kernel_launch(void* const* d_in, const int* in_sizes, int n_in,
                              void* d_out, int out_size, void* d_ws, size_t ws_size,
                              hipStream_t stream) {
  (void)in_sizes; (void)n_in; (void)out_size; (void)ws_size;
  const float* emb  = (const float*)d_in[0];
  const float* W    = (const float*)d_in[1];
  const float* t_in = (const float*)d_in[2];
  const int*   lab  = (const int*)d_in[3];

  float* out        = (float*)d_out;
  float* out2       = out + (size_t)B_ROWS * C_COLS;       // origin_cos * S
  float* scalar_out = out + 2 * (size_t)B_ROWS * C_COLS;   // center loss scalar

  float* w    = (float*)d_ws;
  float* rinv = w;          // 512
  float* ctm  = w + 512;    // 512
  float* ftl  = w + 1024;   // 512
  float* tnew = w + 1536;   // 1

  k_row_norms<<<B_ROWS, 256, 0, stream>>>(emb, rinv);
  k_gemm<<<C_COLS / BN, 256, 0, stream>>>(W, emb, rinv, out2);
  k_target<<<1, 512, 0, stream>>>(out2, lab, t_in, ctm, ftl, tnew, scalar_out);
  k_epilogue<<<(int)(((size_t)B_ROWS * C_COLS) / 4 / 256), 256, 0, stream>>>(
      out2, lab, ctm, ftl, tnew, out);
}